// BoundaryAwareCrossEntropyLoss_86560770884299
// MI455X (gfx1250) — compile-verified
//
#include <hip/hip_runtime.h>
#include <stdint.h>

// Problem constants (from reference: input [8,19,512,1024] f32, target [8,512,1024] i32)
#define Bn 8
#define Cn 19
#define Hn 512
#define Wn 1024
#define HWn (Hn * Wn)        // 524288 = 2^19
#define Nn (Bn * HWn)        // 4194304 pixels
#define NWORDS (Nn / 32)     // 131072 packed bitmap words
#define WPR (Wn / 32)        // 32 words per image row

typedef __attribute__((ext_vector_type(2))) float v2f;
typedef __attribute__((ext_vector_type(8))) float v8f;
typedef __attribute__((ext_vector_type(4))) unsigned int v4u;
typedef __attribute__((ext_vector_type(8))) int v8i;
typedef __attribute__((ext_vector_type(4))) int v4i;

// ---------------------------------------------------------------------------
// Wave32 sum via V_WMMA_F32_16X16X4_F32 (fp32 matrix pipe, no precision loss).
// A(16x4): lane l<16 -> {A[l][0]=p, A[l][1]=0}; lane l>=16 -> {A[l-16][2]=p, [3]=0}.
// B(4x16) = all ones  =>  D[m][n] = p_m + p_{m+16} for every n.
// Lane sums its 8 D VGPRs (covers M=0..7 or M=8..15), shfl_xor(16) completes.
// Requires EXEC all-ones: callers use full blocks with no early return.
// ---------------------------------------------------------------------------
__device__ __forceinline__ float wave_sum32(float v) {
  v2f a; a[0] = v;   a[1] = 0.0f;
  v2f b; b[0] = 1.0f; b[1] = 1.0f;
  v8f c = {0.f, 0.f, 0.f, 0.f, 0.f, 0.f, 0.f, 0.f};
  c = __builtin_amdgcn_wmma_f32_16x16x4_f32(false, a, false, b, (short)0, c,
                                            false, false);
  float s = ((c[0] + c[1]) + (c[2] + c[3])) + ((c[4] + c[5]) + (c[6] + c[7]));
  s += __shfl_xor(s, 16, 32);
  return s;
}

// Two simultaneous block reductions (<=8 waves / 256 threads). Result valid on
// thread 0. Ends with a barrier so shared scratch may be reused immediately.
__device__ __forceinline__ void block_sum2(float v0, float v1, float* out0,
                                           float* out1, float* sm /*16 f*/) {
  float w0 = wave_sum32(v0);
  float w1 = wave_sum32(v1);
  const int lane = threadIdx.x & 31;
  const int wv   = threadIdx.x >> 5;
  if (lane == 0) { sm[wv] = w0; sm[8 + wv] = w1; }
  __syncthreads();
  if (threadIdx.x == 0) {
    float a = 0.f, b = 0.f;
    const int nw = blockDim.x >> 5;
    for (int i = 0; i < nw; ++i) { a += sm[i]; b += sm[8 + i]; }
    *out0 = a; *out1 = b;
  }
  __syncthreads();
}

// ---------------------------------------------------------------------------
// K1: streaming log-softmax NLL, float4-vectorized: each thread owns 4
// contiguous pixels, so each of the 19 channel reads is one global_load_b128
// (512B contiguous per wave per channel). One pass over the 319MB tensor ->
// this kernel is the HBM roofline term (~14us at 23.3 TB/s).
// Target-class logit is re-read as a scalar load (guaranteed cache hit on the
// just-fetched line) instead of a 19-deep cndmask ladder.
// ---------------------------------------------------------------------------
__global__ void nll_kernel(const float* __restrict__ logits,
                           const int* __restrict__ target,
                           float* __restrict__ nll,
                           float* __restrict__ ce_part,
                           float* __restrict__ cnt_part) {
  __shared__ float sm[16];
  const int tid = blockIdx.x * blockDim.x + threadIdx.x;  // 0 .. Nn/4-1
  const int p0  = tid * 4;                                // first of 4 pixels
  const int b   = p0 >> 19;
  const int hw  = p0 & (HWn - 1);                         // multiple of 4
  const float* lbase = logits + (size_t)(b * Cn) * HWn + hw;
  const float4* base4 = (const float4*)lbase;
  __builtin_prefetch((const void*)(lbase + (size_t)Cn * HWn), 0, 0);

  float4 v[Cn];
  float m0 = -3.402823e38f, m1 = m0, m2 = m0, m3 = m0;
#pragma unroll
  for (int c = 0; c < Cn; ++c) {
    v[c] = base4[(size_t)c * (HWn / 4)];
    m0 = fmaxf(m0, v[c].x); m1 = fmaxf(m1, v[c].y);
    m2 = fmaxf(m2, v[c].z); m3 = fmaxf(m3, v[c].w);
  }
  float se0 = 0.f, se1 = 0.f, se2 = 0.f, se3 = 0.f;
#pragma unroll
  for (int c = 0; c < Cn; ++c) {
    se0 += __expf(v[c].x - m0); se1 += __expf(v[c].y - m1);
    se2 += __expf(v[c].z - m2); se3 += __expf(v[c].w - m3);
  }
  const int4 t4 = *(const int4*)(target + p0);
  const bool va0 = (t4.x != 255), va1 = (t4.y != 255);
  const bool va2 = (t4.z != 255), va3 = (t4.w != 255);
  const int c0 = va0 ? t4.x : 0, c1 = va1 ? t4.y : 0;
  const int c2 = va2 ? t4.z : 0, c3 = va3 ? t4.w : 0;
  const float xt0 = lbase[(size_t)c0 * HWn + 0];   // cache-hit reloads
  const float xt1 = lbase[(size_t)c1 * HWn + 1];
  const float xt2 = lbase[(size_t)c2 * HWn + 2];
  const float xt3 = lbase[(size_t)c3 * HWn + 3];
  const float l0 = va0 ? (m0 + __logf(se0) - xt0) : 0.f;
  const float l1 = va1 ? (m1 + __logf(se1) - xt1) : 0.f;
  const float l2 = va2 ? (m2 + __logf(se2) - xt2) : 0.f;
  const float l3 = va3 ? (m3 + __logf(se3) - xt3) : 0.f;
  *(float4*)(nll + p0) = make_float4(l0, l1, l2, l3);

  const float acc = ((l0 + l1) + (l2 + l3));
  const float cnt = (va0 ? 1.f : 0.f) + (va1 ? 1.f : 0.f) +
                    (va2 ? 1.f : 0.f) + (va3 ? 1.f : 0.f);
  block_sum2(acc, cnt, &ce_part[blockIdx.x], &cnt_part[blockIdx.x], sm);
}

// ---------------------------------------------------------------------------
// K2: Canny front half in LDS. Tile 32x8 output; img halo 2 (edge-clamped),
// mag halo 1 (zero outside image). Interior tiles fetch their 36x12 int32
// halo tile via the Tensor Data Mover (one DMA per block, D# per ISA ch.8,
// TENSORcnt-waited); border tiles (which need edge-clamping the TDM's
// rectangular tile cannot express) use manual clamped loads.
// Emits bit-packed weak/strong masks via __ballot (wave == one 32px row).
// ---------------------------------------------------------------------------
__global__ void canny_kernel(const int* __restrict__ target,
                             unsigned int* __restrict__ weak_bits,
                             unsigned int* __restrict__ edge_bits) {
  __shared__ int   ti[12 * 36];     // raw target tile (row-major 12x36)
  __shared__ float img[12][36];
  __shared__ float mg[10][34];
  const int x0 = blockIdx.x * 32;
  const int y0 = blockIdx.y * 8;
  const int b  = blockIdx.z;
  const int lt = threadIdx.y * 32 + threadIdx.x;
  const int* tbase = target + b * HWn;

  const bool interior = (blockIdx.x > 0) && (blockIdx.x < gridDim.x - 1) &&
                        (blockIdx.y > 0) && (blockIdx.y < gridDim.y - 1);
#if __has_builtin(__builtin_amdgcn_tensor_load_to_lds) && \
    __has_builtin(__builtin_amdgcn_s_wait_tensorcnt)
  if (interior) {
    if (lt == 0) {  // one wave issues the DMA (TDM ignores EXEC)
      const int* gsrc = tbase + (y0 - 2) * Wn + (x0 - 2);
      const unsigned long long ga = (unsigned long long)(size_t)gsrc;
      const unsigned int lds_off = (unsigned int)(size_t)&ti[0];
      v4u g0;
      g0[0] = 1u;                                            // count=1
      g0[1] = lds_off;                                       // lds_addr
      g0[2] = (unsigned int)ga;                              // global_addr lo
      g0[3] = (unsigned int)((ga >> 32) & 0x01FFFFFFu) | (2u << 30); // +type=2
      v8i g1;
      g1[0] = (int)(2u << 16);        // wg_mask=0, data_size=4B, no pad/iter
      g1[1] = 0;                      // atomic_barrier_addr=0, td0[15:0]=0
      g1[2] = 16;                     // td0[31:16]=16 (dim0 = 1M, no OOB)
      g1[3] = 16 | (36 << 16);        // td1[31:16]=16 (1M), tile_dim0=36
      g1[4] = 12;                     // tile_dim1=12, tile_dim2=0
      g1[5] = Wn;                     // tensor_dim0_stride = 1024 elements
      g1[6] = 0;                      // stride hi, td1_stride lo
      g1[7] = 0;
      v4i gz4 = {0, 0, 0, 0};         // groups 2/3 unused (2-D tensor)
      v8i gz8 = {0, 0, 0, 0, 0, 0, 0, 0};  // extra group (clang-23 6-arg form)
      __builtin_amdgcn_tensor_load_to_lds(g0, g1, gz4, gz4, gz8, 0);
      __builtin_amdgcn_s_wait_tensorcnt((short)0);
    }
  } else
#endif
  {
    for (int idx = lt; idx < 12 * 36; idx += 256) {
      const int ly = idx / 36, lx = idx - ly * 36;
      int gy = y0 + ly - 2; gy = gy < 0 ? 0 : (gy > Hn - 1 ? Hn - 1 : gy);
      int gx = x0 + lx - 2; gx = gx < 0 ? 0 : (gx > Wn - 1 ? Wn - 1 : gx);
      ti[idx] = tbase[gy * Wn + gx];
    }
  }
  __syncthreads();

  for (int idx = lt; idx < 12 * 36; idx += 256) {
    const int ly = idx / 36, lx = idx - ly * 36;
    img[ly][lx] = (float)((ti[idx] * 255) & 255);  // (t*255) % 256
  }
  __syncthreads();

  for (int idx = lt; idx < 10 * 34; idx += 256) {
    const int my = idx / 34, mx = idx - my * 34;
    const int gy = y0 + my - 1;
    const int gx = x0 + mx - 1;
    float v = 0.f;
    if (gy >= 0 && gy < Hn && gx >= 0 && gx < Wn) {
      const float gxx = (img[my][mx + 2] + 2.f * img[my + 1][mx + 2] + img[my + 2][mx + 2])
                      - (img[my][mx]     + 2.f * img[my + 1][mx]     + img[my + 2][mx]);
      const float gyy = (img[my + 2][mx] + 2.f * img[my + 2][mx + 1] + img[my + 2][mx + 2])
                      - (img[my][mx]     + 2.f * img[my][mx + 1]     + img[my][mx + 2]);
      v = fabsf(gxx) + fabsf(gyy);            // L1 magnitude
    }
    mg[my][mx] = v;
  }
  __syncthreads();

  const int tx = threadIdx.x, ty = threadIdx.y;
  const float gxx = (img[ty + 1][tx + 3] + 2.f * img[ty + 2][tx + 3] + img[ty + 3][tx + 3])
                  - (img[ty + 1][tx + 1] + 2.f * img[ty + 2][tx + 1] + img[ty + 3][tx + 1]);
  const float gyy = (img[ty + 3][tx + 1] + 2.f * img[ty + 3][tx + 2] + img[ty + 3][tx + 3])
                  - (img[ty + 1][tx + 1] + 2.f * img[ty + 1][tx + 2] + img[ty + 1][tx + 3]);
  const float m0 = mg[ty + 1][tx + 1];
  const float ax = fabsf(gxx), ay = fabsf(gyy);
  float n1, n2;
  if (ay <= ax * 0.41421356f)      { n1 = mg[ty + 1][tx];     n2 = mg[ty + 1][tx + 2]; }
  else if (ay >= ax * 2.41421356f) { n1 = mg[ty][tx + 1];     n2 = mg[ty + 2][tx + 1]; }
  else if (gxx * gyy >= 0.f)       { n1 = mg[ty][tx];         n2 = mg[ty + 2][tx + 2]; }
  else                             { n1 = mg[ty][tx + 2];     n2 = mg[ty + 2][tx];     }
  const bool keep = (m0 >= n1) && (m0 > n2);
  const bool wk = keep && (m0 > 50.f);
  const bool st = keep && (m0 > 150.f);
  const unsigned long long wb = __ballot(wk);
  const unsigned long long sb = __ballot(st);
  if (threadIdx.x == 0) {
    const int widx = b * (Hn * WPR) + (y0 + ty) * WPR + blockIdx.x;
    weak_bits[widx] = (unsigned int)wb;
    edge_bits[widx] = (unsigned int)sb;
  }
}

// ---------------------------------------------------------------------------
// K3: hysteresis on bit-packed masks. 32x32 output tile with 16-halo region
// held as 64 uint64 rows in LDS. 16 Jacobi dilation steps per launch; growth
// e <- weak & dilate3x3(e) is monotone, so stale halos are safe. 16 launches
// give the reference's 256-step cap.
// ---------------------------------------------------------------------------
__global__ void hyst_kernel(const unsigned int* __restrict__ weak_bits,
                            unsigned int* __restrict__ edge_bits) {
  __shared__ unsigned long long e[64];
  __shared__ unsigned long long w[64];
  const int y  = threadIdx.x;                 // block = 64 threads, one per row
  const int tx = blockIdx.x, ty = blockIdx.y, b = blockIdx.z;
  const int gy = ty * 32 - 16 + y;
  unsigned long long ew = 0ull, ww = 0ull;
  if (gy >= 0 && gy < Hn) {
    const unsigned int* wr = weak_bits + (size_t)b * Hn * WPR + (size_t)gy * WPR;
    const unsigned int* er = edge_bits + (size_t)b * Hn * WPR + (size_t)gy * WPR;
    const unsigned long long wl = (tx > 0)       ? wr[tx - 1] : 0u;
    const unsigned long long wc =                  wr[tx];
    const unsigned long long wg = (tx < WPR - 1) ? wr[tx + 1] : 0u;
    ww = (wl >> 16) | (wc << 16) | (wg << 48);  // 64-bit window at col tx*32-16
    const unsigned long long el = (tx > 0)       ? er[tx - 1] : 0u;
    const unsigned long long ec =                  er[tx];
    const unsigned long long eg = (tx < WPR - 1) ? er[tx + 1] : 0u;
    ew = (el >> 16) | (ec << 16) | (eg << 48);
  }
  w[y] = ww; e[y] = ew;
  __syncthreads();
#pragma unroll 1
  for (int it = 0; it < 16; ++it) {
    const unsigned long long up = (y > 0)  ? e[y - 1] : 0ull;
    const unsigned long long dn = (y < 63) ? e[y + 1] : 0ull;
    unsigned long long m = e[y] | up | dn;
    m = m | (m << 1) | (m >> 1);              // 3x3 dilation, zero-padded
    const unsigned long long ne = w[y] & m;
    __syncthreads();
    e[y] = ne;
    __syncthreads();
  }
  if (y >= 16 && y < 48) {                    // central 32 rows x 32 cols
    edge_bits[(size_t)b * Hn * WPR + (size_t)gy * WPR + tx] =
        (unsigned int)(e[y] >> 16);
  }
}

// ---------------------------------------------------------------------------
// K4: boundary-weighted sum: iterate set bits of the final edge bitmap,
// gather nll, reduce per block (WMMA wave reduction, full blocks).
// ---------------------------------------------------------------------------
__global__ void bsum_kernel(const unsigned int* __restrict__ edge_bits,
                            const float* __restrict__ nll,
                            float* __restrict__ bsum_part,
                            float* __restrict__ bcnt_part) {
  __shared__ float sm[16];
  const int widx = blockIdx.x * blockDim.x + threadIdx.x;  // 0 .. NWORDS-1
  unsigned int bits = edge_bits[widx];
  const float* p = nll + (size_t)widx * 32;
  float s = 0.f, c = 0.f;
  while (bits) {
    const int bpos = __ffs(bits) - 1;
    s += p[bpos];
    c += 1.f;
    bits &= bits - 1;
  }
  block_sum2(s, c, &bsum_part[blockIdx.x], &bcnt_part[blockIdx.x], sm);
}

// ---------------------------------------------------------------------------
// K5: final single-block reduction and loss assembly (deterministic order).
// ---------------------------------------------------------------------------
__global__ void final_kernel(const float* __restrict__ ce_part,
                             const float* __restrict__ cnt_part,
                             const float* __restrict__ bsum_part,
                             const float* __restrict__ bcnt_part,
                             float* __restrict__ out) {
  __shared__ float sm[16];
  __shared__ float res[4];
  float ce = 0.f, cnt = 0.f;
  for (int i = threadIdx.x; i < 4096; i += 256) { ce += ce_part[i]; cnt += cnt_part[i]; }
  float bs = 0.f, bc = 0.f;
  for (int i = threadIdx.x; i < 512; i += 256) { bs += bsum_part[i]; bc += bcnt_part[i]; }
  block_sum2(ce, cnt, &res[0], &res[1], sm);
  block_sum2(bs, bc, &res[2], &res[3], sm);
  if (threadIdx.x == 0) {
    const float cem = res[0] / fmaxf(res[1], 1.f);
    const float bm  = (res[3] > 0.f) ? 10.f * (res[2] / fmaxf(res[3], 1.f)) : 0.f;
    out[0] = cem + bm;
  }
}

extern "C" void kernel_launch(void* const* d_in, const int* in_sizes, int n_in,
                              void* d_out, int out_size, void* d_ws, size_t ws_size,
                              hipStream_t stream) {
  (void)in_sizes; (void)n_in; (void)out_size; (void)ws_size;
  const float* logits = (const float*)d_in[0];   // [8,19,512,1024] f32
  const int*   target = (const int*)d_in[1];     // [8,512,1024]    i32
  float* out = (float*)d_out;                    // scalar loss

  // Workspace carve-up (~17.9 MB): nll | weak bitmap | edge bitmap | partials
  char* ws = (char*)d_ws;
  float*        nll       = (float*)ws;                                   // 16 MB
  unsigned int* weak_bits = (unsigned int*)(ws + (size_t)Nn * 4);          // 512 KB
  unsigned int* edge_bits = (unsigned int*)(ws + (size_t)Nn * 4 + (size_t)NWORDS * 4);
  float*        ce_part   = (float*)(ws + (size_t)Nn * 4 + (size_t)NWORDS * 8);
  float*        cnt_part  = ce_part + 4096;
  float*        bsum_part = cnt_part + 4096;
  float*        bcnt_part = bsum_part + 512;

  nll_kernel<<<Nn / 4 / 256, 256, 0, stream>>>(logits, target, nll, ce_part, cnt_part);
  canny_kernel<<<dim3(Wn / 32, Hn / 8, Bn), dim3(32, 8, 1), 0, stream>>>(
      target, weak_bits, edge_bits);
  for (int it = 0; it < 16; ++it)
    hyst_kernel<<<dim3(Wn / 32, Hn / 32, Bn), 64, 0, stream>>>(weak_bits, edge_bits);
  bsum_kernel<<<NWORDS / 256, 256, 0, stream>>>(edge_bits, nll, bsum_part, bcnt_part);
  final_kernel<<<1, 256, 0, stream>>>(ce_part, cnt_part, bsum_part, bcnt_part, out);
}